// Net_30940944400971
// MI455X (gfx1250) — compile-verified
//
#include <hip/hip_runtime.h>

// LIF spiking net: Dense(1) GEMV via V_WMMA_F32_16X16X4_F32, then sequential scan.
//
// Roofline: 262 MB of x @ 23.3 TB/s ~= 11 us floor; 131 MFLOP is negligible.
// f32 WMMA keeps bit-exact-ish accumulation (threshold compares are sensitive),
// and the matrix pipe is nowhere near the bottleneck at this intensity.

typedef __attribute__((ext_vector_type(2))) float v2f;
typedef __attribute__((ext_vector_type(4))) float v4f;
typedef __attribute__((ext_vector_type(8))) float v8f;

#define BATCH   256
#define TSTEPS  2000
#define FEAT    128
#define NROWS   (BATCH * TSTEPS)        // 512000 dot products
#define NTILES  (NROWS / 16)            // 32000 16-row WMMA tiles
#define BETA    0.8807970779778823f     // sigmoid(2.0)
#define VTH     1.0f

// ---------------------------------------------------------------------------
// Phase 1: cur[r] = dot(x[r, 0:128], W) + b   for r in [0, 512000)
// One wave per 16-row tile. f32 WMMA 16x16x4, K accumulated over 32 steps.
//
// A-fragment layout (ISA 7.12.2, 32-bit A 16x4): lane L<16 -> row M=L holds
// K={k0,k0+1}; lane L+16 -> row M=L holds K={k0+2,k0+3}. B(4x16) mirrors the
// K split per lane-half; we broadcast W over all 16 columns so every column
// of D carries the same dot product -> read column N=0 from lanes 0 and 16.
// ---------------------------------------------------------------------------
__global__ __launch_bounds__(256) void lif_dense_wmma(
    const float* __restrict__ x,
    const float* __restrict__ W,
    const float* __restrict__ bias,
    float*       __restrict__ cur)
{
    const int lane = threadIdx.x & 31;
    const int wave = threadIdx.x >> 5;
    const int tile = blockIdx.x * 8 + wave;        // exactly NTILES tiles
    const long rowbase = (long)tile * 16;

    const int m   = lane & 15;                     // matrix row within tile
    const int sel = lane >> 4;                     // 0: K+{0,1}, 1: K+{2,3}

    const float bv = bias[0];
    v8f c = {bv, bv, bv, bv, bv, bv, bv, bv};      // seed accumulator with bias

    const float* __restrict__ xrow = x + (rowbase + m) * FEAT + 2 * sel;
    const float* __restrict__ wrow = W + 2 * sel;

#pragma unroll
    for (int k0 = 0; k0 < FEAT; k0 += 4) {
        v2f a = *(const v2f*)(xrow + k0);          // x[row, k0+2sel .. +1]
        v2f w = *(const v2f*)(wrow + k0);          // W[k0+2sel .. +1] (bcast over N)
        // 8 args: (neg_a, A, neg_b, B, c_mod, C, reuse_a, reuse_b)
        c = __builtin_amdgcn_wmma_f32_16x16x4_f32(
                false, a, false, w, (short)0, c, false, false);
    }

    // Column N=0: lane 0 holds rows 0..7 in c[0..7], lane 16 holds rows 8..15.
    if (m == 0) {
        float* dst = cur + rowbase + sel * 8;
        v4f lo = {c[0], c[1], c[2], c[3]};
        v4f hi = {c[4], c[5], c[6], c[7]};
        *(v4f*)(dst)     = lo;
        *(v4f*)(dst + 4) = hi;
    }
}

// ---------------------------------------------------------------------------
// Phase 2: per-batch-row LIF scan, in place on d_out.
//   v = BETA*v + c; s = (v - VTH >= 0); v -= s*VTH
// Reads cur[b*T + t], overwrites with spike at the same index (single thread
// owns the whole row -> safe in-place; avoids relying on d_ws sizing).
// Total traffic 4 MB, all L2-resident; latency chain is 4 VALU ops per step.
// ---------------------------------------------------------------------------
__global__ __launch_bounds__(32) void lif_scan(float* __restrict__ out)
{
    const int b = blockIdx.x * blockDim.x + threadIdx.x;
    if (b >= BATCH) return;

    float* __restrict__ row = out + (long)b * TSTEPS;
    float v = 0.0f;

    for (int t = 0; t < TSTEPS; t += 4) {
        v4f c4 = *(v4f*)(row + t);
#pragma unroll
        for (int j = 0; j < 4; ++j) {
            v = fmaf(BETA, v, c4[j]);              // leaky integrate
            float s = (v >= VTH) ? 1.0f : 0.0f;    // heaviside(v - VTH)
            v -= s * VTH;                          // subtraction reset
            c4[j] = s;
        }
        *(v4f*)(row + t) = c4;
    }
    out[NROWS + b] = v;                            // final membrane voltage
}

// ---------------------------------------------------------------------------
extern "C" void kernel_launch(void* const* d_in, const int* in_sizes, int n_in,
                              void* d_out, int out_size, void* d_ws, size_t ws_size,
                              hipStream_t stream)
{
    (void)in_sizes; (void)n_in; (void)out_size; (void)d_ws; (void)ws_size;

    const float* x    = (const float*)d_in[0];     // [256, 2000, 128]
    const float* W    = (const float*)d_in[1];     // [128, 1]
    const float* bias = (const float*)d_in[2];     // [1]
    float*       out  = (float*)d_out;             // [512000 spikes | 256 vT]

    // 32000 tiles / 8 waves-per-block = 4000 blocks of 256 threads (8 wave32).
    lif_dense_wmma<<<NTILES / 8, 256, 0, stream>>>(x, W, bias, out);
    // 256 independent scans, one lane each.
    lif_scan<<<BATCH / 32, 32, 0, stream>>>(out);
}